// GNNWithAttention_22170621182629
// MI455X (gfx1250) — compile-verified
//
#include <hip/hip_runtime.h>
#include <cmath>

// Problem constants (from reference)
#define N_NODES 50000
#define N_EDGES 800000
#define N_EP    850000   // E + N self loops
#define DH      128
#define NGRAPH  64
#define EPSF    1e-5f

typedef float v2f __attribute__((ext_vector_type(2)));
typedef float v8f __attribute__((ext_vector_type(8)));

// LDS layout for W: keyed by k-pair kp = k>>1 (64 kp rows).
//   lds[kp * KPSTRIDE + col*2 + (k&1)]
// -> the two K values a lane feeds to WMMA are one contiguous 8B word
//    (single ds_load_b64, even-aligned, no VGPR shuffling).
// KPSTRIDE = 288 dwords: odd/even lane-half bases differ by 288 % 64 == 32
// banks -> lanes 0-15 and 16-31 use disjoint bank halves (conflict-free).
#define KPSTRIDE 288

// ---------------------------------------------------------------------------
// fp32 WMMA GEMM: C[n x 128] = A[n x 128] * W[128 x 128]
// Block = 256 threads (8 waves). Each block computes 128 rows; each wave a
// 16-row x 128-col strip via 32 x 8 V_WMMA_F32_16X16X4_F32.
// ---------------------------------------------------------------------------
__global__ __launch_bounds__(256) void gemm_nodes_wmma(
    const float* __restrict__ A, const float* __restrict__ W,
    float* __restrict__ C, int n) {
  __shared__ float lw[64 * KPSTRIDE];  // 72KB (WGP has 320KB)
  const int tid = threadIdx.x;
  // Stage W (128x128 f32) into LDS, coalesced float4 global reads.
  for (int i = tid; i < DH * (DH / 4); i += 256) {
    const int k  = i >> 5;             // 32 float4 per row
    const int c0 = (i & 31) << 2;
    const float4 v = reinterpret_cast<const float4*>(W)[i];
    float* p = &lw[(k >> 1) * KPSTRIDE + c0 * 2 + (k & 1)];
    p[0] = v.x; p[2] = v.y; p[4] = v.z; p[6] = v.w;
  }
  __syncthreads();

  const int wave = tid >> 5;
  const int lane = tid & 31;
  const int row0 = blockIdx.x * 128 + wave * 16;
  if (row0 >= n) return;               // after the only barrier: safe

  const int half = lane >> 4;          // K split across lane halves
  const int l    = lane & 15;

  v8f acc[8] = {};                     // 8 col-tiles x 8 VGPRs = 128-wide strip

  const float* arow = A + (size_t)(row0 + l) * DH + 2 * half;
  for (int kb = 0; kb < 32; ++kb) {    // K = 128 in steps of 4
    // A[M=l][K = kb*4 + 2*half + {0,1}] : one global_load_b64
    const v2f a = *reinterpret_cast<const v2f*>(arow + kb * 4);
    // B-fragment base for this lane: kp = kb*2 + half, column l
    const float* bbase = lw + (kb * 2 + half) * KPSTRIDE + l * 2;
#pragma unroll
    for (int t = 0; t < 8; ++t) {      // 8 column tiles of 16
      // W[K][t*16+l], W[K+1][t*16+l] packed contiguously: one ds_load_b64
      const v2f b = *reinterpret_cast<const v2f*>(bbase + t * 32);
      acc[t] = __builtin_amdgcn_wmma_f32_16x16x4_f32(
          false, a, false, b, (short)0, acc[t], false, false);
    }
  }
#pragma unroll
  for (int t = 0; t < 8; ++t) {
#pragma unroll
    for (int v = 0; v < 8; ++v) {      // C/D layout: VGPR v -> M = v + 8*half
      C[(size_t)(row0 + v + 8 * half) * DH + t * 16 + l] = acc[t][v];
    }
  }
}

// ---------------------------------------------------------------------------
// Per-node attention logits: als[r] = h[r,:]·a_src, ald[r] = h[r,:]·a_dst
// wave32 per row, float4 per lane, shfl_xor reduction.
// ---------------------------------------------------------------------------
__global__ __launch_bounds__(256) void node_attn_logits(
    const float* __restrict__ h, const float* __restrict__ a_src,
    const float* __restrict__ a_dst, float* __restrict__ als,
    float* __restrict__ ald, int n) {
  const int wave = threadIdx.x >> 5, lane = threadIdx.x & 31;
  const int row = blockIdx.x * 8 + wave;
  if (row >= n) return;
  const float4 hv = reinterpret_cast<const float4*>(h + (size_t)row * DH)[lane];
  const float4 as = reinterpret_cast<const float4*>(a_src)[lane];
  const float4 ad = reinterpret_cast<const float4*>(a_dst)[lane];
  float s = hv.x * as.x + hv.y * as.y + hv.z * as.z + hv.w * as.w;
  float d = hv.x * ad.x + hv.y * ad.y + hv.z * ad.z + hv.w * ad.w;
  for (int off = 16; off; off >>= 1) {
    s += __shfl_xor(s, off, 32);
    d += __shfl_xor(d, off, 32);
  }
  if (lane == 0) { als[row] = s; ald[row] = d; }
}

// float atomic max via signed/unsigned integer atomics (IEEE ordering trick)
__device__ __forceinline__ void atomicMaxF32(float* addr, float v) {
  if (v >= 0.0f) atomicMax(reinterpret_cast<int*>(addr), __float_as_int(v));
  else atomicMin(reinterpret_cast<unsigned int*>(addr), __float_as_uint(v));
}

// Pass A: alpha = leaky_relu(als[src]+ald[dst]); m[dst] = max(...)
__global__ __launch_bounds__(256) void edge_alpha_max(
    const int* __restrict__ src, const int* __restrict__ dst,
    const float* __restrict__ als, const float* __restrict__ ald,
    float* __restrict__ ew, float* __restrict__ m, int nedge, int ebase) {
  const int e = blockIdx.x * 256 + threadIdx.x;
  if (e >= nedge) return;
  int s, d;
  if (e < ebase) { s = src[e]; d = dst[e]; } else { s = d = e - ebase; }
  float a = als[s] + ald[d];
  a = a > 0.0f ? a : 0.2f * a;      // leaky_relu slope 0.2
  ew[e] = a;
  atomicMaxF32(&m[d], a);
}

// Pass B: e = exp(alpha - m[dst]); z[dst] += e
__global__ __launch_bounds__(256) void edge_exp_sum(
    const int* __restrict__ dst, float* __restrict__ ew,
    const float* __restrict__ m, float* __restrict__ z, int nedge, int ebase) {
  const int e = blockIdx.x * 256 + threadIdx.x;
  if (e >= nedge) return;
  const int d = (e < ebase) ? dst[e] : (e - ebase);
  const float ex = expf(ew[e] - m[d]);
  ew[e] = ex;
  atomicAdd(&z[d], ex);
}

// Pass C: out[dst] += (e/z[dst]) * h[src]; wave32 per edge, float4 per lane.
__global__ __launch_bounds__(256) void edge_scatter(
    const int* __restrict__ src, const int* __restrict__ dst,
    const float* __restrict__ h, const float* __restrict__ ew,
    const float* __restrict__ z, float* __restrict__ out,
    int nedge, int ebase) {
  const int wave = threadIdx.x >> 5, lane = threadIdx.x & 31;
  const int e = blockIdx.x * 8 + wave;
  if (e >= nedge) return;
  int s, d;
  if (e < ebase) { s = src[e]; d = dst[e]; } else { s = d = e - ebase; }
  const float w = ew[e] / z[d];
  const float4 hv = reinterpret_cast<const float4*>(h + (size_t)s * DH)[lane];
  float* o = out + (size_t)d * DH + lane * 4;
  atomicAdd(o + 0, w * hv.x);
  atomicAdd(o + 1, w * hv.y);
  atomicAdd(o + 2, w * hv.z);
  atomicAdd(o + 3, w * hv.w);
}

// bias (+ optional relu) -> layernorm -> optional relu ; wave32 per row
__global__ __launch_bounds__(256) void node_finalize(
    const float* __restrict__ acc, const float* __restrict__ bias,
    const float* __restrict__ g, const float* __restrict__ b,
    float* __restrict__ out, int n, int relu_before, int relu_after) {
  const int wave = threadIdx.x >> 5, lane = threadIdx.x & 31;
  const int row = blockIdx.x * 8 + wave;
  if (row >= n) return;
  float4 v = reinterpret_cast<const float4*>(acc + (size_t)row * DH)[lane];
  const float4 bb = reinterpret_cast<const float4*>(bias)[lane];
  v.x += bb.x; v.y += bb.y; v.z += bb.z; v.w += bb.w;
  if (relu_before) {
    v.x = fmaxf(v.x, 0.0f); v.y = fmaxf(v.y, 0.0f);
    v.z = fmaxf(v.z, 0.0f); v.w = fmaxf(v.w, 0.0f);
  }
  float s = v.x + v.y + v.z + v.w;
  for (int off = 16; off; off >>= 1) s += __shfl_xor(s, off, 32);
  const float mu = s * (1.0f / DH);
  float4 c = {v.x - mu, v.y - mu, v.z - mu, v.w - mu};
  float q = c.x * c.x + c.y * c.y + c.z * c.z + c.w * c.w;
  for (int off = 16; off; off >>= 1) q += __shfl_xor(q, off, 32);
  const float rstd = rsqrtf(q * (1.0f / DH) + EPSF);
  const float4 gg = reinterpret_cast<const float4*>(g)[lane];
  const float4 be = reinterpret_cast<const float4*>(b)[lane];
  float4 r;
  r.x = c.x * rstd * gg.x + be.x;
  r.y = c.y * rstd * gg.y + be.y;
  r.z = c.z * rstd * gg.z + be.z;
  r.w = c.w * rstd * gg.w + be.w;
  if (relu_after) {
    r.x = fmaxf(r.x, 0.0f); r.y = fmaxf(r.y, 0.0f);
    r.z = fmaxf(r.z, 0.0f); r.w = fmaxf(r.w, 0.0f);
  }
  reinterpret_cast<float4*>(out + (size_t)row * DH)[lane] = r;
}

// global mean pool: atomic accumulate per graph
__global__ __launch_bounds__(256) void pool_accum(
    const float* __restrict__ h, const int* __restrict__ batch,
    float* __restrict__ pooled, float* __restrict__ cnts, int n) {
  const int wave = threadIdx.x >> 5, lane = threadIdx.x & 31;
  const int row = blockIdx.x * 8 + wave;
  if (row >= n) return;
  const int gi = batch[row];
  const float4 hv = reinterpret_cast<const float4*>(h + (size_t)row * DH)[lane];
  float* p = pooled + gi * DH + lane * 4;
  atomicAdd(p + 0, hv.x); atomicAdd(p + 1, hv.y);
  atomicAdd(p + 2, hv.z); atomicAdd(p + 3, hv.w);
  if (lane == 0) atomicAdd(&cnts[gi], 1.0f);
}

__global__ __launch_bounds__(256) void pool_div(float* pooled,
                                                const float* cnts) {
  const int i = blockIdx.x * 256 + threadIdx.x;  // NGRAPH*DH total
  const float c = fmaxf(cnts[i >> 7], 1.0f);
  pooled[i] /= c;
}

// logits = pooled @ fc_W + fc_b  (64 x 128 @ 128 x 128, trivially small)
__global__ __launch_bounds__(128) void fc_kernel(
    const float* __restrict__ pooled, const float* __restrict__ Wf,
    const float* __restrict__ bf, float* __restrict__ logits) {
  const int g = blockIdx.x, o = threadIdx.x;
  float s = bf[o];
  const float* pr = pooled + g * DH;
  for (int k = 0; k < DH; ++k) s += pr[k] * Wf[k * DH + o];
  logits[g * DH + o] = s;
}

// BatchNorm1d over batch dim (G=64 rows), training-mode batch stats
__global__ __launch_bounds__(128) void bn_kernel(
    const float* __restrict__ logits, const float* __restrict__ g,
    const float* __restrict__ b, float* __restrict__ out) {
  const int j = threadIdx.x;
  float mu = 0.0f;
  for (int r = 0; r < NGRAPH; ++r) mu += logits[r * DH + j];
  mu *= (1.0f / NGRAPH);
  float var = 0.0f;
  for (int r = 0; r < NGRAPH; ++r) {
    const float c = logits[r * DH + j] - mu;
    var += c * c;
  }
  var *= (1.0f / NGRAPH);
  const float rs = rsqrtf(var + EPSF);
  const float gj = g[j], bj = b[j];
  for (int r = 0; r < NGRAPH; ++r)
    out[r * DH + j] = (logits[r * DH + j] - mu) * rs * gj + bj;
}

__global__ void fill_f32(float* p, float v, long n) {
  const long i = (long)blockIdx.x * blockDim.x + threadIdx.x;
  if (i < n) p[i] = v;
}

// ---------------------------------------------------------------------------
extern "C" void kernel_launch(void* const* d_in, const int* in_sizes, int n_in,
                              void* d_out, int out_size, void* d_ws,
                              size_t ws_size, hipStream_t stream) {
  (void)in_sizes; (void)n_in; (void)out_size; (void)ws_size;
  // inputs in setup_inputs() order
  const float* x      = (const float*)d_in[0];
  const int*   eidx   = (const int*)d_in[1];   // [2, E] flattened: row0=src, row1=dst
  const int*   batch  = (const int*)d_in[2];
  const float* W1     = (const float*)d_in[3];
  const float* a_src1 = (const float*)d_in[4];
  const float* a_dst1 = (const float*)d_in[5];
  const float* b1     = (const float*)d_in[6];
  const float* W2     = (const float*)d_in[7];
  const float* a_src2 = (const float*)d_in[8];
  const float* a_dst2 = (const float*)d_in[9];
  const float* b2     = (const float*)d_in[10];
  const float* ln1_g  = (const float*)d_in[11];
  const float* ln1_b  = (const float*)d_in[12];
  const float* ln2_g  = (const float*)d_in[13];
  const float* ln2_b  = (const float*)d_in[14];
  const float* fc_W   = (const float*)d_in[15];
  const float* fc_b   = (const float*)d_in[16];
  const float* bn_g   = (const float*)d_in[17];
  const float* bn_b   = (const float*)d_in[18];

  const int* srcp = eidx;
  const int* dstp = eidx + N_EDGES;

  // workspace layout (floats)
  float* ws = (float*)d_ws;
  const size_t nd = (size_t)N_NODES * DH;
  float* hA     = ws;                 // node feature buffer A
  float* hB     = hA + nd;            // node feature buffer B / accumulator
  float* als    = hB + nd;
  float* ald    = als + N_NODES;
  float* mbuf   = ald + N_NODES;
  float* zbuf   = mbuf + N_NODES;
  float* ew     = zbuf + N_NODES;     // per-edge scratch (E')
  float* pooled = ew + N_EP;
  float* cnts   = pooled + NGRAPH * DH;
  float* logits = cnts + NGRAPH;

  const int gemm_grid = (N_NODES + 127) / 128;          // 391
  const int row_grid  = (N_NODES + 7) / 8;              // 6250
  const int edg_grid  = (N_EP + 255) / 256;             // 3321
  const int edw_grid  = (N_EP + 7) / 8;                 // 106250

  // ---------------- Layer 1 ----------------
  gemm_nodes_wmma<<<gemm_grid, 256, 0, stream>>>(x, W1, hA, N_NODES);
  node_attn_logits<<<row_grid, 256, 0, stream>>>(hA, a_src1, a_dst1, als, ald,
                                                 N_NODES);
  fill_f32<<<(N_NODES + 255) / 256, 256, 0, stream>>>(mbuf, -INFINITY, N_NODES);
  fill_f32<<<(N_NODES + 255) / 256, 256, 0, stream>>>(zbuf, 0.0f, N_NODES);
  fill_f32<<<(int)((nd + 255) / 256), 256, 0, stream>>>(hB, 0.0f, (long)nd);
  edge_alpha_max<<<edg_grid, 256, 0, stream>>>(srcp, dstp, als, ald, ew, mbuf,
                                               N_EP, N_EDGES);
  edge_exp_sum<<<edg_grid, 256, 0, stream>>>(dstp, ew, mbuf, zbuf, N_EP,
                                             N_EDGES);
  edge_scatter<<<edw_grid, 256, 0, stream>>>(srcp, dstp, hA, ew, zbuf, hB,
                                             N_EP, N_EDGES);
  node_finalize<<<row_grid, 256, 0, stream>>>(hB, b1, ln1_g, ln1_b, hA,
                                              N_NODES, /*relu_before=*/1,
                                              /*relu_after=*/0);

  // ---------------- Layer 2 ----------------
  gemm_nodes_wmma<<<gemm_grid, 256, 0, stream>>>(hA, W2, hB, N_NODES);
  node_attn_logits<<<row_grid, 256, 0, stream>>>(hB, a_src2, a_dst2, als, ald,
                                                 N_NODES);
  fill_f32<<<(N_NODES + 255) / 256, 256, 0, stream>>>(mbuf, -INFINITY, N_NODES);
  fill_f32<<<(N_NODES + 255) / 256, 256, 0, stream>>>(zbuf, 0.0f, N_NODES);
  fill_f32<<<(int)((nd + 255) / 256), 256, 0, stream>>>(hA, 0.0f, (long)nd);
  edge_alpha_max<<<edg_grid, 256, 0, stream>>>(srcp, dstp, als, ald, ew, mbuf,
                                               N_EP, N_EDGES);
  edge_exp_sum<<<edg_grid, 256, 0, stream>>>(dstp, ew, mbuf, zbuf, N_EP,
                                             N_EDGES);
  edge_scatter<<<edw_grid, 256, 0, stream>>>(srcp, dstp, hB, ew, zbuf, hA,
                                             N_EP, N_EDGES);
  node_finalize<<<row_grid, 256, 0, stream>>>(hA, b2, ln2_g, ln2_b, hB,
                                              N_NODES, /*relu_before=*/0,
                                              /*relu_after=*/1);

  // ---------------- Pool + FC + BN ----------------
  fill_f32<<<(NGRAPH * DH + NGRAPH + 255) / 256, 256, 0, stream>>>(
      pooled, 0.0f, (long)(NGRAPH * DH + NGRAPH));
  pool_accum<<<row_grid, 256, 0, stream>>>(hB, batch, pooled, cnts, N_NODES);
  pool_div<<<(NGRAPH * DH) / 256, 256, 0, stream>>>(pooled, cnts);
  fc_kernel<<<NGRAPH, 128, 0, stream>>>(pooled, fc_W, fc_b, logits);
  bn_kernel<<<1, 128, 0, stream>>>(logits, bn_g, bn_b, (float*)d_out);
}